// ClusterFPS_58437325029838
// MI455X (gfx1250) — compile-verified
//
#include <hip/hip_runtime.h>
#include <hip/hip_bf16.h>
#include <stddef.h>

// ---------------------------------------------------------------------------
// ClusterFPS pipeline for gfx1250 (MI455X).
// B=8, N=65536, D=3, K=64 clusters, NITER=10, M=128 neighbors, S=64 samples.
// k-means assignment einsum mapped onto V_WMMA_F32_16X16X4_F32 (wave32).
// All argmin/argmax selections use packed 64-bit ordered keys -> branchless
// v_cmp_lt_u64 + v_cndmask instead of exec-mask branches.
// ---------------------------------------------------------------------------

typedef __attribute__((ext_vector_type(2))) float v2f;
typedef __attribute__((ext_vector_type(8))) float v8f;
typedef unsigned long long u64;

#define NB   8
#define NPTS 65536
#define NC   64      // clusters / centers
#define NM   128     // neighbors per center
#define NS   64      // FPS samples per group
#define NIT  10

// float -> order-preserving unsigned bits (handles negatives)
__device__ __forceinline__ unsigned orderedBits(float s) {
  unsigned u = __float_as_uint(s);
  return u ^ ((unsigned)(((int)u) >> 31) | 0x80000000u);
}
// packed key: primary = float order, secondary = index (min => argmin, low idx)
__device__ __forceinline__ u64 packKey(float s, unsigned n) {
  return ((u64)orderedBits(s) << 32) | (u64)n;
}

// ---------------------------------------------------------------------------
// init: centers = x[:, :64, :], accumulators = 0
// ---------------------------------------------------------------------------
__global__ void kmeans_init(const float* __restrict__ x,
                            float* __restrict__ centers,
                            float* __restrict__ acc) {
  int i = blockIdx.x * blockDim.x + threadIdx.x;   // 0..511
  if (i < NB * NC) {
    int b = i >> 6, n = i & 63;
    const float* p = x + ((size_t)b * NPTS + n) * 3;
    centers[i * 3 + 0] = p[0];
    centers[i * 3 + 1] = p[1];
    centers[i * 3 + 2] = p[2];
    acc[i * 4 + 0] = 0.f; acc[i * 4 + 1] = 0.f;
    acc[i * 4 + 2] = 0.f; acc[i * 4 + 3] = 0.f;
  }
}

// ---------------------------------------------------------------------------
// assignment + accumulation. Block = 256 threads = 8 waves, each wave does
// 16 tiles of 16 points -> 2048 points/block. Grid = (32, 8).
// Per 16-pt tile: 4x V_WMMA_F32_16X16X4_F32 (A = 16x4 points [x,y,z,0],
// B = 4x16 centers) -> dot[m][n]; argmin_n(|c_n|^2 - 2 dot).
// A layout (32-bit 16x4): lanes 0-15 hold K=0,1 ; lanes 16-31 hold K=2,3.
// C/D layout: lane l holds col n=l%16, VGPR v holds row m=v+8*(l/16).
// ---------------------------------------------------------------------------
__global__ __launch_bounds__(256)
void kmeans_assign(const float* __restrict__ x,
                   const float* __restrict__ centers,
                   float* __restrict__ acc) {
  const int b    = blockIdx.y;
  const int tid  = threadIdx.x;
  const int lane = tid & 31;
  const int wave = tid >> 5;
  const int half = lane >> 4;   // 0: K=0,1 side, 1: K=2,3 side
  const int lm   = lane & 15;   // row (A) / col (B,D) within tile

  __shared__ float lds_acc[NC * 4];
  for (int i = tid; i < NC * 4; i += 256) lds_acc[i] = 0.0f;
  __syncthreads();

  const float* xb = x + (size_t)b * NPTS * 3;
  const float* cb = centers + b * NC * 3;

  // B fragments (centers, column-major 4x16) + |c|^2 per own column, 4 tiles
  v2f  bfrag[4];
  float c2t[4];
#pragma unroll
  for (int t = 0; t < 4; ++t) {
    int n = t * 16 + lm;
    float cx = cb[n * 3 + 0], cy = cb[n * 3 + 1], cz = cb[n * 3 + 2];
    bfrag[t][0] = half ? cz : cx;        // K=2 : K=0
    bfrag[t][1] = half ? 0.0f : cy;      // K=3 : K=1
    c2t[t] = cx * cx + cy * cy + cz * cz;
  }

  const int base = blockIdx.x * 2048 + wave * 256;

  for (int tile = 0; tile < 16; ++tile) {
    const int p0   = base + tile * 16;
    const int pIdx = p0 + lm;
    float px = xb[pIdx * 3 + 0];
    float py = xb[pIdx * 3 + 1];
    float pz = xb[pIdx * 3 + 2];

    v2f afrag;
    afrag[0] = half ? pz : px;
    afrag[1] = half ? 0.0f : py;

    v8f cz8 = {};
    v8f d0 = __builtin_amdgcn_wmma_f32_16x16x4_f32(false, afrag, false, bfrag[0],
                                                   (short)0, cz8, false, false);
    v8f d1 = __builtin_amdgcn_wmma_f32_16x16x4_f32(false, afrag, false, bfrag[1],
                                                   (short)0, cz8, false, false);
    v8f d2 = __builtin_amdgcn_wmma_f32_16x16x4_f32(false, afrag, false, bfrag[2],
                                                   (short)0, cz8, false, false);
    v8f d3 = __builtin_amdgcn_wmma_f32_16x16x4_f32(false, afrag, false, bfrag[3],
                                                   (short)0, cz8, false, false);

#pragma unroll
    for (int v = 0; v < 8; ++v) {
      // packed candidates from the 4 center tiles; min => smallest score,
      // tie -> smallest center index (matches jnp.argmin)
      u64 best = packKey(c2t[0] - 2.0f * d0[v], (unsigned)(lm));
      u64 k;
      k = packKey(c2t[1] - 2.0f * d1[v], (unsigned)(16 + lm)); best = (k < best) ? k : best;
      k = packKey(c2t[2] - 2.0f * d2[v], (unsigned)(32 + lm)); best = (k < best) ? k : best;
      k = packKey(c2t[3] - 2.0f * d3[v], (unsigned)(48 + lm)); best = (k < best) ? k : best;
      // branchless min across the 16 lanes of this half (row m's 16 columns)
#pragma unroll
      for (int off = 1; off < 16; off <<= 1) {
        u64 o = __shfl_xor(best, off, 32);
        best = (o < best) ? o : best;
      }
      if (lm == v) {                      // one writer per (half, row)
        int bestN = (int)(unsigned)(best & 0xffffffffull);
        int m  = v + 8 * half;            // row handled by this half's VGPR v
        int pw = p0 + m;
        float wx = xb[pw * 3 + 0];
        float wy = xb[pw * 3 + 1];
        float wz = xb[pw * 3 + 2];
        atomicAdd(&lds_acc[bestN * 4 + 0], wx);
        atomicAdd(&lds_acc[bestN * 4 + 1], wy);
        atomicAdd(&lds_acc[bestN * 4 + 2], wz);
        atomicAdd(&lds_acc[bestN * 4 + 3], 1.0f);
      }
    }
  }

  __syncthreads();
  float* accB = acc + b * NC * 4;
  for (int i = tid; i < NC * 4; i += 256) atomicAdd(&accB[i], lds_acc[i]);
}

// ---------------------------------------------------------------------------
// update: centers = sum / count ; re-zero accumulators; mirror into d_out tail
// ---------------------------------------------------------------------------
__global__ void kmeans_update(float* __restrict__ acc,
                              float* __restrict__ centers,
                              float* __restrict__ centers_out) {
  int i = blockIdx.x * blockDim.x + threadIdx.x;   // 0..511
  if (i < NB * NC) {
    float cnt = acc[i * 4 + 3];
    float inv = 1.0f / cnt;
    float cx = acc[i * 4 + 0] * inv;
    float cy = acc[i * 4 + 1] * inv;
    float cz = acc[i * 4 + 2] * inv;
    centers[i * 3 + 0] = cx; centers[i * 3 + 1] = cy; centers[i * 3 + 2] = cz;
    centers_out[i * 3 + 0] = cx; centers_out[i * 3 + 1] = cy; centers_out[i * 3 + 2] = cz;
    acc[i * 4 + 0] = 0.f; acc[i * 4 + 1] = 0.f;
    acc[i * 4 + 2] = 0.f; acc[i * 4 + 3] = 0.f;
  }
}

// ---------------------------------------------------------------------------
// top-128 nearest points per (b, center): 4-round radix select on float-bit
// keys (dist >= 0 -> bits are order-preserving), then bitonic sort of 128
// packed (distBits<<32 | idx) keys. One block (256 thr) per query; x stays
// L2-resident across the 5 passes.
// ---------------------------------------------------------------------------
__global__ __launch_bounds__(256)
void query_topk(const float* __restrict__ x,
                const float* __restrict__ centers,
                float* __restrict__ neigh) {
  const int n   = blockIdx.x;   // center
  const int b   = blockIdx.y;   // batch
  const int tid = threadIdx.x;

  const float* xb = x + (size_t)b * NPTS * 3;
  const float* c  = centers + (b * NC + n) * 3;
  const float cx = c[0], cy = c[1], cz = c[2];

  __shared__ unsigned hist[256];
  __shared__ unsigned s_prefix;
  __shared__ int      s_want;
  __shared__ int      s_cntLess, s_cntEq;
  __shared__ u64      cdi[NM];

  if (tid == 0) { s_prefix = 0u; s_want = NM; }
  __syncthreads();

  for (int round = 0; round < 4; ++round) {
    hist[tid] = 0u;
    __syncthreads();
    const unsigned prefix = s_prefix;
    const int shift = 24 - 8 * round;
    for (int i = tid; i < NPTS; i += 256) {
      float dx = xb[i * 3 + 0] - cx;
      float dy = xb[i * 3 + 1] - cy;
      float dz = xb[i * 3 + 2] - cz;
      unsigned key = __float_as_uint(dx * dx + dy * dy + dz * dz);
      if (round == 0 || (key >> (shift + 8)) == prefix)
        atomicAdd(&hist[(key >> shift) & 255u], 1u);
    }
    __syncthreads();
    if (tid == 0) {
      int want = s_want;
      unsigned cum = 0;
      int sel = 255;
      for (int bin = 0; bin < 256; ++bin) {
        unsigned h = hist[bin];
        if (cum + h >= (unsigned)want) { sel = bin; break; }
        cum += h;
      }
      s_want   = want - (int)cum;
      s_prefix = (prefix << 8) | (unsigned)sel;
    }
    __syncthreads();
  }

  const unsigned T = s_prefix;       // exact key of the 128th smallest
  const int nTies = s_want;          // # of ==T elements to keep
  const int lessTotal = NM - nTies;  // # of <T elements (exact)
  if (tid == 0) { s_cntLess = 0; s_cntEq = 0; }
  __syncthreads();

  for (int i = tid; i < NPTS; i += 256) {
    float dx = xb[i * 3 + 0] - cx;
    float dy = xb[i * 3 + 1] - cy;
    float dz = xb[i * 3 + 2] - cz;
    unsigned key = __float_as_uint(dx * dx + dy * dy + dz * dz);
    u64 pk = ((u64)key << 32) | (unsigned)i;
    if (key < T) {
      int p = atomicAdd(&s_cntLess, 1);
      cdi[p] = pk;
    } else if (key == T) {
      int p = atomicAdd(&s_cntEq, 1);
      if (p < nTies) cdi[lessTotal + p] = pk;
    }
  }
  __syncthreads();

  // bitonic sort of the 128 packed keys, ascending (dist, then idx)
  for (int size = 2; size <= NM; size <<= 1) {
    for (int stride = size >> 1; stride > 0; stride >>= 1) {
      __syncthreads();
      if (tid < NM) {
        int j = tid ^ stride;
        if (j > tid) {
          u64 a  = cdi[tid];
          u64 bb = cdi[j];
          bool up = ((tid & size) == 0);
          u64 lo = (a < bb) ? a : bb;
          u64 hi = (a < bb) ? bb : a;
          cdi[tid] = up ? lo : hi;
          cdi[j]   = up ? hi : lo;
        }
      }
    }
  }
  __syncthreads();

  if (tid < NM) {
    int pi = (int)(unsigned)(cdi[tid] & 0xffffffffull);
    float* o = neigh + (((size_t)b * NC + n) * NM + tid) * 3;
    o[0] = xb[pi * 3 + 0];
    o[1] = xb[pi * 3 + 1];
    o[2] = xb[pi * 3 + 2];
  }
}

// ---------------------------------------------------------------------------
// FPS: one 128-thread block per (b, group); 64 steps. Sample s writes the
// farthest index *before* the update (matches jax.lax.scan carry output).
// Argmax uses packed keys: (distBits<<32) | (127 - tid) so max => first index
// on ties (matches jnp.argmax), reduced branchlessly in LDS.
// ---------------------------------------------------------------------------
__global__ __launch_bounds__(128)
void fps_kernel(const float* __restrict__ neigh,
                const int* __restrict__ far0,
                float* __restrict__ out) {
  const int g   = blockIdx.x;
  const int b   = blockIdx.y;
  const int tid = threadIdx.x;   // one point per thread

  const float* pts = neigh + ((size_t)b * NC + g) * NM * 3;
  const float px = pts[tid * 3 + 0];
  const float py = pts[tid * 3 + 1];
  const float pz = pts[tid * 3 + 2];

  float dist = 1e10f;
  int far = far0[b * NC + g];

  __shared__ float cen[3];
  __shared__ u64   rk[NM];

  float* ob = out + (((size_t)b * NC + g) * NS) * 3;

  for (int s = 0; s < NS; ++s) {
    if (tid == far) {
      cen[0] = px; cen[1] = py; cen[2] = pz;
      ob[s * 3 + 0] = px; ob[s * 3 + 1] = py; ob[s * 3 + 2] = pz;
    }
    __syncthreads();
    float dx = px - cen[0], dy = py - cen[1], dz = pz - cen[2];
    dist = fminf(dist, dx * dx + dy * dy + dz * dz);
    // dist >= 0 -> raw float bits are order-preserving
    rk[tid] = ((u64)__float_as_uint(dist) << 32) | (unsigned)(NM - 1 - tid);
    __syncthreads();
#pragma unroll
    for (int off = NM / 2; off > 0; off >>= 1) {
      if (tid < off) {
        u64 o = rk[tid + off];
        u64 m = rk[tid];
        rk[tid] = (o > m) ? o : m;
      }
      __syncthreads();
    }
    far = NM - 1 - (int)(unsigned)(rk[0] & 0xffffffffull);
    __syncthreads();
  }
}

// ---------------------------------------------------------------------------
// launch
// ---------------------------------------------------------------------------
extern "C" void kernel_launch(void* const* d_in, const int* in_sizes, int n_in,
                              void* d_out, int out_size, void* d_ws, size_t ws_size,
                              hipStream_t stream) {
  const float* x    = (const float*)d_in[0];   // (8, 65536, 3) f32
  const int*   far0 = (const int*)d_in[1];     // (8, 64) i32
  float* out = (float*)d_out;                  // new_xyz 98304 f32 || centers 1536 f32
  float* ws  = (float*)d_ws;

  float* centers = ws;                         // 1536 f32
  float* acc     = ws + NB * NC * 3;           // 2048 f32
  float* neigh   = acc + NB * NC * 4;          // 196608 f32

  float* centers_out = out + NB * NC * NS * 3; // tail of d_out (8*64*64*3 = 98304)

  kmeans_init<<<2, 256, 0, stream>>>(x, centers, acc);
  for (int it = 0; it < NIT; ++it) {
    kmeans_assign<<<dim3(32, NB), 256, 0, stream>>>(x, centers, acc);
    kmeans_update<<<2, 256, 0, stream>>>(acc, centers, centers_out);
  }
  query_topk<<<dim3(NC, NB), 256, 0, stream>>>(x, centers, neigh);
  fps_kernel<<<dim3(NC, NB), 128, 0, stream>>>(neigh, far0, out);

  (void)in_sizes; (void)n_in; (void)out_size; (void)ws_size;
}